// SNNLayer_1958505087425
// MI455X (gfx1250) — compile-verified
//
#include <hip/hip_runtime.h>
#include <hip/hip_bf16.h>
#include <math.h>

// Problem constants from the reference
#define B_DIM 64
#define O_DIM 512
#define N_DIM 784
#define NPAD  1024
#define BIG_F 1e10f

// d_ws layout (bytes):
//   [0,          200704)  s_ws   : B*N floats  (sorted exp values)
//   [200704,     401408)  idx_ws : B*N ints    (argsort indices)
//   [401408,    2007040)  wT     : N*O floats  (transposed weights)
#define S_OFF   0
#define IDX_OFF 200704
#define WT_OFF  401408

// ---------------------------------------------------------------------------
// Kernel 0: tiled transpose w[O,N] -> wT[N,O] so the hot-loop gather
// wT[c*512 + o] is coalesced across the 512 lanes (16 wave32s) of a block.
// ---------------------------------------------------------------------------
__global__ void snn_transpose_w(const float* __restrict__ w,
                                float* __restrict__ wT) {
    __shared__ float tile[32][33];           // +1 pad: no LDS bank conflicts
    const int n0 = blockIdx.x * 32;
    const int o0 = blockIdx.y * 32;
    const int tx = threadIdx.x, ty = threadIdx.y;

    const int n = n0 + tx;                   // coalesced read along N
    const int o = o0 + ty;                   // o < 512 always (grid.y*32 == 512)
    if (n < N_DIM) tile[ty][tx] = w[o * N_DIM + n];
    __syncthreads();

    const int o2 = o0 + tx;                  // coalesced write along O
    const int n2 = n0 + ty;
    if (n2 < N_DIM) wT[n2 * O_DIM + o2] = tile[tx][ty];
}

// ---------------------------------------------------------------------------
// Kernel 1: per-batch-row bitonic sort (ascending) of e = exp(1.79*x) with
// index tracking, fully in LDS. 1024-padded with +inf so pads sort to the top.
// One block per batch row, 512 threads each handle one compare-exchange/stage.
// ---------------------------------------------------------------------------
__global__ void snn_sort_rows(const float* __restrict__ x,
                              float* __restrict__ s_ws,
                              int* __restrict__ idx_ws) {
    __shared__ float key[NPAD];
    __shared__ int   kid[NPAD];
    const int b = blockIdx.x;
    const int t = threadIdx.x;               // 0..511

    for (int i = t; i < NPAD; i += 512) {
        float v = (i < N_DIM) ? __expf(1.79f * x[b * N_DIM + i])
                              : __builtin_inff();
        key[i] = v;
        kid[i] = i;
    }
    __syncthreads();

    for (int k = 2; k <= NPAD; k <<= 1) {
        for (int j = k >> 1; j > 0; j >>= 1) {
            const int i = ((t & ~(j - 1)) << 1) | (t & (j - 1));
            const int p = i | j;
            const bool asc = ((i & k) == 0);
            float a = key[i], c = key[p];
            if (asc ? (a > c) : (a < c)) {
                key[i] = c; key[p] = a;
                int ti = kid[i]; kid[i] = kid[p]; kid[p] = ti;
            }
            __syncthreads();
        }
    }

    for (int i = t; i < N_DIM; i += 512) {
        s_ws[b * N_DIM + i]   = key[i];
        idx_ws[b * N_DIM + i] = kid[i];
    }
}

// ---------------------------------------------------------------------------
// Kernel 2: the faithful first-hit scan. One block per batch row b; thread t
// owns output column o = t. s[b,:] and idx[b,:] are staged into LDS with
// CDNA5 async global->LDS b128 copies (ASYNCcnt path), then each lane scans
// j = 0..783 gathering wT[idx[j]*512 + t] (coalesced), reproducing:
//   ws  = wg[j] + wg[j-1], wim = mul[j] + mul[j-1]
//   oa  = wim / clip(ws-1, 1e-10, 1e10)
//   hit = (oa < s[j-1] (inf at j=0)) && (ws > 1)
//   out = first-hit oa, else oa at j=0 (argmax of all-False is 0).
// ---------------------------------------------------------------------------
__global__ void snn_scan(const float* __restrict__ wT,
                         const float* __restrict__ s_ws,
                         const int* __restrict__ idx_ws,
                         float* __restrict__ out) {
    __shared__ __align__(16) float s_sh[N_DIM];
    __shared__ __align__(16) int   i_sh[N_DIM];
    const int b = blockIdx.x;
    const int t = threadIdx.x;               // output column o

    // ---- async stage: 784 floats = 196 x b128 per array ----
    if (t < 196) {
        unsigned lds_s = (unsigned)(unsigned long long)(&s_sh[t * 4]);
        unsigned lds_i = (unsigned)(unsigned long long)(&i_sh[t * 4]);
        unsigned long long ga_s =
            (unsigned long long)(const void*)(s_ws + (size_t)b * N_DIM + t * 4);
        unsigned long long ga_i =
            (unsigned long long)(const void*)(idx_ws + (size_t)b * N_DIM + t * 4);
        asm volatile(
            "global_load_async_to_lds_b128 %0, %1, off\n\t"
            "global_load_async_to_lds_b128 %2, %3, off"
            :
            : "v"(lds_s), "v"(ga_s), "v"(lds_i), "v"(ga_i)
            : "memory");
    }
    asm volatile("s_wait_asynccnt 0" ::: "memory");
    __syncthreads();

    // ---- scan ----
    float pw = 0.0f;                 // wg[j-1]   (0 so j=0 sums reduce to wg0)
    float pm = 0.0f;                 // mul[j-1]
    float pins = __builtin_inff();   // inp_shift: s[j-1], inf at j=0
    float out0 = 0.0f;               // out_all at j=0 (argmax-of-false fallback)
    float hitv = 0.0f;
    bool  found = false;

    for (int j = 0; j < N_DIM; ++j) {
        const int c = i_sh[j];
        // prefetch a future gather column into cache (global_prefetch_b8)
        const int pf = (j + 16 < N_DIM) ? (j + 16) : (N_DIM - 1);
        __builtin_prefetch(&wT[(size_t)i_sh[pf] * O_DIM + t], 0, 3);

        const float wg  = wT[(size_t)c * O_DIM + t];   // coalesced across lanes
        const float sj  = s_sh[j];
        const float mul = sj * wg;
        const float ws  = wg + pw;
        const float wim = mul + pm;
        float den = ws - 1.0f;
        den = fminf(fmaxf(den, 1e-10f), BIG_F);
        const float oa = wim / den;
        if (j == 0) out0 = oa;
        if ((oa < pins) && (ws > 1.0f)) { hitv = oa; found = true; break; }
        pw = wg; pm = mul; pins = sj;
    }
    // Appended column N can never fire (BIG < s[N-1] is false), so the
    // all-false case takes index 0 per jnp.argmax semantics.
    out[(size_t)b * O_DIM + t] = found ? hitv : out0;
}

// ---------------------------------------------------------------------------
extern "C" void kernel_launch(void* const* d_in, const int* in_sizes, int n_in,
                              void* d_out, int out_size, void* d_ws, size_t ws_size,
                              hipStream_t stream) {
    const float* x = (const float*)d_in[0];   // [64, 784]
    const float* w = (const float*)d_in[1];   // [512, 784]
    float* out = (float*)d_out;               // [64, 512]

    float* s_ws   = (float*)((char*)d_ws + S_OFF);
    int*   idx_ws = (int*)((char*)d_ws + IDX_OFF);
    float* wT     = (float*)((char*)d_ws + WT_OFF);

    snn_transpose_w<<<dim3(25, 16), dim3(32, 32), 0, stream>>>(w, wT);
    snn_sort_rows<<<B_DIM, 512, 0, stream>>>(x, s_ws, idx_ws);
    snn_scan<<<B_DIM, 512, 0, stream>>>(wT, s_ws, idx_ws, out);
}